// GeometricEnergyAttention_29678224016079
// MI455X (gfx1250) — compile-verified
//
#include <hip/hip_runtime.h>
#include <hip/hip_bf16.h>
#include <math.h>

// ---------------- constants from the reference ----------------
#define Bn        2
#define Ln        1024
#define Mn        32
#define F_DIM     256
#define C_DIM     32
#define Hn        12
#define QKn       16
#define HQK       (Hn * QKn)          // 192
#define FEAT_DIM  (Hn * C_DIM + Hn * QKn + 7 * Hn)   // 384 + 192 + 84 = 660
#define NNODES    (Bn * Ln)           // 2048
#define EPS_DIR   1.0e-4f
#define LN_EPS    1.0e-5f
#define BETA_COEF 0.23570226039551584f   // sqrt(2/9)/2
#define LOGIT_SCALE 0.5773502691896258f  // sqrt(1/3)

typedef __attribute__((ext_vector_type(2))) float v2f;
typedef __attribute__((ext_vector_type(8))) float v8f;

// =====================================================================
// Kernel 1: QKV projection.  C[which] = X (2048x256) @ W[which] (256x192)
// One wave per 16x16 fp32 tile, v_wmma_f32_16x16x4_f32, K consumed 4/step.
// B tile (256x16) staged in LDS as k-pair float2 so B frag is ds_load_b64.
// grid = 3 * 12 * 16 = 576 blocks of 256 threads (8 waves -> 8 m-tiles).
// =====================================================================
__global__ __launch_bounds__(256) void qkv_gemm_kernel(
    const float* __restrict__ X,
    const float* __restrict__ Wq, const float* __restrict__ Wk,
    const float* __restrict__ Wv, float* __restrict__ qkv)
{
    __shared__ v2f Bs[(F_DIM / 2) * 16];   // 128 k-pairs x 16 cols = 16 KB

    const int which = blockIdx.x / 192;          // 0=q 1=k 2=v
    const int rem   = blockIdx.x % 192;
    const int tn    = rem % 12;                  // N tile (192/16)
    const int tmg   = rem / 12;                  // group of 8 M tiles
    const float* W  = (which == 0) ? Wq : (which == 1) ? Wk : Wv;

    const int tid = threadIdx.x;
    for (int i = tid; i < (F_DIM / 2) * 16; i += 256) {
        const int kp = i >> 4, nn = i & 15;
        v2f t;
        t.x = W[(size_t)(2 * kp)     * HQK + tn * 16 + nn];
        t.y = W[(size_t)(2 * kp + 1) * HQK + tn * 16 + nn];
        Bs[i] = t;
    }
    __syncthreads();

    const int wave = tid >> 5, lane = tid & 31;
    const int half = lane >> 4, l16 = lane & 15;
    const int tm   = tmg * 8 + wave;
    const int m0   = tm * 16;
    // A frag: lane<16 holds K={k,k+1} of row m0+l16; lane>=16 holds K={k+2,k+3}
    const float* arow = X + (size_t)(m0 + l16) * F_DIM + 2 * half;

    v8f acc = {};
#pragma unroll 8
    for (int k = 0; k < F_DIM; k += 4) {
        v2f a = *(const v2f*)(arow + k);
        v2f b = Bs[((k >> 1) + half) * 16 + l16];
        acc = __builtin_amdgcn_wmma_f32_16x16x4_f32(
            false, a, false, b, (short)0, acc, false, false);
    }

    float* C = qkv + (size_t)which * NNODES * HQK;
    const int col = tn * 16 + l16;
#pragma unroll
    for (int r = 0; r < 8; ++r)
        C[(size_t)(m0 + r + 8 * half) * HQK + col] = acc[r];
}

// =====================================================================
// Kernel 2: per-node gathered attention. One 128-thread block per node.
// Produces feat (660 floats) = [p2n(384) | node(192) | local(36) | dist(12) | dirn(36)]
// =====================================================================
__global__ __launch_bounds__(128) void attn_node_kernel(
    const float* __restrict__ Rm,   const float* __restrict__ tvec,
    const float* __restrict__ pCB,  const float* __restrict__ z,
    const int*   __restrict__ neighbors,
    const float* __restrict__ Wpair, const float* __restrict__ spatial_coef,
    const float* __restrict__ qkv,  float* __restrict__ feat_ws)
{
    const int n   = blockIdx.x;          // node 0..2047
    const int bi  = n >> 10;
    const int li  = n & 1023;
    const int tid = threadIdx.x;

    __shared__ int   nb_s[Mn];
    __shared__ float zk_s[Mn][C_DIM];    // 4 KB gathered pair features
    __shared__ float q_s[HQK];
    __shared__ float pk_s[Mn][3];
    __shared__ float ps_s[3];
    __shared__ float R_s[9];
    __shared__ float t_s[3];
    __shared__ float logit_s[Mn][Hn];
    __shared__ float alpha_s[Mn][Hn];
    __shared__ float aggr_s[Hn][3];
    __shared__ float feat_s[FEAT_DIM];

    const float* q_node = qkv + (size_t)n * HQK;
    const float* k_base = qkv + (size_t)1 * NNODES * HQK;
    const float* v_base = qkv + (size_t)2 * NNODES * HQK;

    if (tid < Mn)                   nb_s[tid]      = neighbors[(size_t)n * Mn + tid];
    else if (tid < Mn + 9)          R_s[tid - 32]  = Rm[(size_t)n * 9 + (tid - 32)];
    else if (tid < Mn + 12)         t_s[tid - 41]  = tvec[(size_t)n * 3 + (tid - 41)];
    else if (tid < Mn + 15)         ps_s[tid - 44] = pCB[(size_t)n * 3 + (tid - 44)];
    __syncthreads();

    {   // gather z rows: 4 threads per neighbor row, 8 floats each (coalesced)
        const int m  = tid >> 2;
        const int c0 = (tid & 3) * 8;
        const float* zr = z + (((size_t)bi * Ln + li) * Ln + nb_s[m]) * C_DIM;
#pragma unroll
        for (int j = 0; j < 8; ++j) zk_s[m][c0 + j] = zr[c0 + j];
    }
    for (int i = tid; i < HQK; i += 128) q_s[i] = q_node[i];
    if (tid < Mn * 3) {
        const int m = tid / 3, i = tid % 3;
        pk_s[m][i] = pCB[((size_t)bi * Ln + nb_s[m]) * 3 + i];
    }
    __syncthreads();

    // logits[m][h] = (q.k + z.Wpair - d2*softplus(coef)*BETA) * SCALE
    for (int p = tid; p < Mn * Hn; p += 128) {
        const int m = p / Hn, h = p % Hn;
        const float* krow = k_base + ((size_t)bi * Ln + nb_s[m]) * HQK + h * QKn;
        float accd = 0.f;
#pragma unroll
        for (int d = 0; d < QKn; ++d) accd += q_s[h * QKn + d] * krow[d];
        float accp = 0.f;
#pragma unroll
        for (int c = 0; c < C_DIM; ++c) accp += zk_s[m][c] * Wpair[c * Hn + h];
        const float dx = ps_s[0] - pk_s[m][0];
        const float dy = ps_s[1] - pk_s[m][1];
        const float dz = ps_s[2] - pk_s[m][2];
        const float d2 = dx * dx + dy * dy + dz * dz;
        const float sc = spatial_coef[h];
        const float gamma = (sc > 20.f) ? sc : log1pf(__expf(sc));
        logit_s[m][h] = (accd + accp - d2 * gamma * BETA_COEF) * LOGIT_SCALE;
    }
    __syncthreads();

    if (tid < Hn) {                       // softmax over m per head
        const int h = tid;
        float mx = -1e30f;
        for (int m = 0; m < Mn; ++m) mx = fmaxf(mx, logit_s[m][h]);
        float s = 0.f;
        for (int m = 0; m < Mn; ++m) {
            const float e = __expf(logit_s[m][h] - mx);
            alpha_s[m][h] = e; s += e;
        }
        const float inv = 1.f / s;
        for (int m = 0; m < Mn; ++m) alpha_s[m][h] *= inv;
    }
    __syncthreads();

    // feat_p2n[h][c]
    for (int p = tid; p < Hn * C_DIM; p += 128) {
        const int h = p >> 5, c = p & 31;
        float a = 0.f;
#pragma unroll
        for (int m = 0; m < Mn; ++m) a += alpha_s[m][h] * zk_s[m][c];
        feat_s[p] = a;
    }
    // feat_node[h][d]  (p == h*16+d indexes v rows directly)
    for (int p = tid; p < HQK; p += 128) {
        const int h = p >> 4;
        float a = 0.f;
        for (int m = 0; m < Mn; ++m)
            a += alpha_s[m][h] * v_base[((size_t)bi * Ln + nb_s[m]) * HQK + p];
        feat_s[Hn * C_DIM + p] = a;
    }
    if (tid < Hn * 3) {                   // aggr[h][i]
        const int h = tid / 3, i = tid % 3;
        float a = 0.f;
        for (int m = 0; m < Mn; ++m) a += alpha_s[m][h] * pk_s[m][i];
        aggr_s[h][i] = a;
    }
    __syncthreads();
    if (tid < Hn) {                       // local, dist, dirn
        const int h = tid;
        float lx[3];
#pragma unroll
        for (int i = 0; i < 3; ++i) {
            float a = 0.f;
#pragma unroll
            for (int j = 0; j < 3; ++j) a += R_s[j * 3 + i] * (aggr_s[h][j] - t_s[j]);
            lx[i] = a;
        }
        const float dist = sqrtf(lx[0]*lx[0] + lx[1]*lx[1] + lx[2]*lx[2]);
        const float inv  = 1.f / (dist + EPS_DIR);
        const int base = Hn * C_DIM + HQK;            // 576
#pragma unroll
        for (int i = 0; i < 3; ++i) feat_s[base + h * 3 + i] = lx[i];
        feat_s[base + 36 + h] = dist;                  // 612..623
#pragma unroll
        for (int i = 0; i < 3; ++i) feat_s[base + 48 + h * 3 + i] = lx[i] * inv; // 624..659
    }
    __syncthreads();
    for (int i = tid; i < FEAT_DIM; i += 128)
        feat_ws[(size_t)n * FEAT_DIM + i] = feat_s[i];
}

// =====================================================================
// Kernel 3: output projection + bias + mask + residual.
// h = x + mask * (feat (2048x660) @ Wout (660x256) + b_out)
// Same WMMA structure; Wout tile (660x16 = 42.2 KB) staged in LDS.
// grid = 16 (tn) * 16 (tm groups) = 256 blocks of 256 threads.
// =====================================================================
__global__ __launch_bounds__(256) void out_gemm_kernel(
    const float* __restrict__ feat, const float* __restrict__ Wout,
    const float* __restrict__ b_out, const float* __restrict__ X,
    const unsigned char* __restrict__ mask, float* __restrict__ hbuf)
{
    __shared__ v2f Bs[(FEAT_DIM / 2) * 16];   // 330 k-pairs x 16 = 42.2 KB

    const int tn  = blockIdx.x & 15;
    const int tmg = blockIdx.x >> 4;
    const int tid = threadIdx.x;

    for (int i = tid; i < (FEAT_DIM / 2) * 16; i += 256) {
        const int kp = i >> 4, nn = i & 15;
        v2f t;
        t.x = Wout[(size_t)(2 * kp)     * F_DIM + tn * 16 + nn];
        t.y = Wout[(size_t)(2 * kp + 1) * F_DIM + tn * 16 + nn];
        Bs[i] = t;
    }
    __syncthreads();

    const int wave = tid >> 5, lane = tid & 31;
    const int half = lane >> 4, l16 = lane & 15;
    const int m0   = (tmg * 8 + wave) * 16;
    const float* arow = feat + (size_t)(m0 + l16) * FEAT_DIM + 2 * half;

    v8f acc = {};
#pragma unroll 5
    for (int k = 0; k < FEAT_DIM; k += 4) {
        v2f a = *(const v2f*)(arow + k);
        v2f b = Bs[((k >> 1) + half) * 16 + l16];
        acc = __builtin_amdgcn_wmma_f32_16x16x4_f32(
            false, a, false, b, (short)0, acc, false, false);
    }

    const int col = tn * 16 + l16;
    const float bo = b_out[col];
#pragma unroll
    for (int r = 0; r < 8; ++r) {
        const int row = m0 + r + 8 * half;
        float v = acc[r] + bo;
        if (!mask[row]) v = 0.f;
        hbuf[(size_t)row * F_DIM + col] = X[(size_t)row * F_DIM + col] + v;
    }
}

// =====================================================================
// Kernel 4: LayerNorm over F_DIM=256, one wave32 per row (8 elems/lane).
// =====================================================================
__global__ __launch_bounds__(256) void layernorm_kernel(
    const float* __restrict__ hbuf, const float* __restrict__ gamma,
    const float* __restrict__ beta, float* __restrict__ out)
{
    const int wave = threadIdx.x >> 5, lane = threadIdx.x & 31;
    const int row  = blockIdx.x * 8 + wave;
    const float* hr = hbuf + (size_t)row * F_DIM;

    const float4 v0 = *(const float4*)(hr + lane * 8);
    const float4 v1 = *(const float4*)(hr + lane * 8 + 4);
    float s  = v0.x + v0.y + v0.z + v0.w + v1.x + v1.y + v1.z + v1.w;
    float ss = v0.x*v0.x + v0.y*v0.y + v0.z*v0.z + v0.w*v0.w
             + v1.x*v1.x + v1.y*v1.y + v1.z*v1.z + v1.w*v1.w;
#pragma unroll
    for (int off = 16; off > 0; off >>= 1) {
        s  += __shfl_xor(s,  off, 32);
        ss += __shfl_xor(ss, off, 32);
    }
    const float mu  = s * (1.f / F_DIM);
    const float var = ss * (1.f / F_DIM) - mu * mu;
    const float inv = rsqrtf(var + LN_EPS);

    float* orow = out + (size_t)row * F_DIM;
    const int c0 = lane * 8;
    float4 o0, o1;
    o0.x = (v0.x - mu) * inv * gamma[c0+0] + beta[c0+0];
    o0.y = (v0.y - mu) * inv * gamma[c0+1] + beta[c0+1];
    o0.z = (v0.z - mu) * inv * gamma[c0+2] + beta[c0+2];
    o0.w = (v0.w - mu) * inv * gamma[c0+3] + beta[c0+3];
    o1.x = (v1.x - mu) * inv * gamma[c0+4] + beta[c0+4];
    o1.y = (v1.y - mu) * inv * gamma[c0+5] + beta[c0+5];
    o1.z = (v1.z - mu) * inv * gamma[c0+6] + beta[c0+6];
    o1.w = (v1.w - mu) * inv * gamma[c0+7] + beta[c0+7];
    *(float4*)(orow + c0)     = o0;
    *(float4*)(orow + c0 + 4) = o1;
}

// =====================================================================
extern "C" void kernel_launch(void* const* d_in, const int* in_sizes, int n_in,
                              void* d_out, int out_size, void* d_ws, size_t ws_size,
                              hipStream_t stream) {
    const float* R            = (const float*)d_in[0];
    const float* t            = (const float*)d_in[1];
    const float* p_CB         = (const float*)d_in[2];
    const float* x            = (const float*)d_in[3];
    const float* z            = (const float*)d_in[4];
    const unsigned char* mask = (const unsigned char*)d_in[5];
    const int*   neighbors    = (const int*)d_in[6];
    const float* Wq           = (const float*)d_in[7];
    const float* Wk           = (const float*)d_in[8];
    const float* Wv           = (const float*)d_in[9];
    const float* Wpair        = (const float*)d_in[10];
    const float* spatial_coef = (const float*)d_in[11];
    const float* Wout         = (const float*)d_in[12];
    const float* b_out        = (const float*)d_in[13];
    const float* ln_gamma     = (const float*)d_in[14];
    const float* ln_beta      = (const float*)d_in[15];
    float* out = (float*)d_out;

    float* ws   = (float*)d_ws;
    float* qkv  = ws;                                           // 3*2048*192
    float* feat = ws + (size_t)3 * NNODES * HQK;                // 2048*660
    float* hbuf = feat + (size_t)NNODES * FEAT_DIM;             // 2048*256

    qkv_gemm_kernel<<<576, 256, 0, stream>>>(x, Wq, Wk, Wv, qkv);
    attn_node_kernel<<<NNODES, 128, 0, stream>>>(R, t, p_CB, z, neighbors,
                                                 Wpair, spatial_coef, qkv, feat);
    out_gemm_kernel<<<256, 256, 0, stream>>>(feat, Wout, b_out, x, mask, hbuf);
    layernorm_kernel<<<NNODES / 8, 256, 0, stream>>>(hbuf, ln_gamma, ln_beta, out);
}